// NSHE_39883066310766
// MI455X (gfx1250) — compile-verified
//
#include <hip/hip_runtime.h>
#include <math.h>

typedef float v2f __attribute__((ext_vector_type(2)));
typedef float v8f __attribute__((ext_vector_type(8)));

#define N_NODES 100000
#define N_EDGES 1600000
#define EDIM    64

// ---------------------------------------------------------------------------
// 1) Init: agg[n][d] = b[d]  (agg lives in d_out's h region), zero logit accum
// ---------------------------------------------------------------------------
__global__ void nshe_init_kernel(float* __restrict__ agg,
                                 const float* __restrict__ b,
                                 float* __restrict__ logit_acc) {
    int i = blockIdx.x * blockDim.x + threadIdx.x;
    if (i < N_NODES * EDIM) agg[i] = b[i & (EDIM - 1)];
    if (i == 0) *logit_acc = 0.0f;
}

// ---------------------------------------------------------------------------
// 2) GEMM y = x @ W via v_wmma_f32_16x16x4_f32.
//    Block = 256 threads = 8 waves. Waves tile 32 rows x 64 cols:
//    wave w -> rows [blk*32 + (w>>2)*16, +16), cols [(w&3)*16, +16).
//    W staged in LDS (pitch 65 to avoid bank conflicts across lane halves).
// ---------------------------------------------------------------------------
__global__ void __launch_bounds__(256)
nshe_gemm_kernel(const float* __restrict__ x,
                 const float* __restrict__ W,
                 float* __restrict__ y) {
    __shared__ float Ws[64 * 65];
    for (int i = threadIdx.x; i < 64 * 64; i += 256) {
        int r = i >> 6, c = i & 63;
        Ws[r * 65 + c] = W[i];
    }
    __syncthreads();

    const int wave = threadIdx.x >> 5;
    const int lane = threadIdx.x & 31;
    const int half = lane >> 4;      // 0: lanes 0-15, 1: lanes 16-31
    const int lm   = lane & 15;
    const int r0   = blockIdx.x * 32 + (wave >> 2) * 16;
    const int c0   = (wave & 3) * 16;

    const float* xrow = x + (size_t)(r0 + lm) * EDIM;

    v8f acc = {};
#pragma unroll
    for (int k0 = 0; k0 < 64; k0 += 4) {
        const int ka = k0 + 2 * half;              // K = vgpr + 2*half
        // A fragment: lane lm holds row r0+lm, K = {ka, ka+1}
        v2f a = *(const v2f*)&xrow[ka];
        // B fragment: lane lm holds col c0+lm, K = {ka, ka+1}
        v2f bf;
        bf.x = Ws[(ka + 0) * 65 + c0 + lm];
        bf.y = Ws[(ka + 1) * 65 + c0 + lm];
        acc = __builtin_amdgcn_wmma_f32_16x16x4_f32(
            /*neg_a=*/false, a, /*neg_b=*/false, bf,
            /*c_mod=*/(short)0, acc, /*reuse_a=*/false, /*reuse_b=*/false);
    }

    // D layout: VGPR j -> row r0 + j + 8*half, col c0 + lm
    float* yout = y + (size_t)r0 * EDIM + c0 + lm;
#pragma unroll
    for (int j = 0; j < 8; ++j)
        yout[(size_t)(j + 8 * half) * EDIM] = acc[j];
}

// ---------------------------------------------------------------------------
// 3) Edge scatter: agg[dst[e]][d] += y[src[e]][d]   (L2-resident atomics)
//    thread t -> edge t>>6, dim t&63 (wave-uniform edge index, coalesced dims)
// ---------------------------------------------------------------------------
__global__ void nshe_scatter_kernel(const float* __restrict__ y,
                                    const int* __restrict__ src,
                                    const int* __restrict__ dst,
                                    float* __restrict__ agg) {
    long long t = (long long)blockIdx.x * blockDim.x + threadIdx.x;
    if (t >= (long long)N_EDGES * EDIM) return;
    int e = (int)(t >> 6);
    int d = (int)(t & 63);
    float val = y[(size_t)src[e] * EDIM + d];
    atomicAdd(&agg[(size_t)dst[e] * EDIM + d], val);
}

// ---------------------------------------------------------------------------
// 4) Fused normalize + classifier partial:
//    wave32 per row; lane handles elems {l, l+32} (scalar b32: h is 4B-aligned).
//    h[n] /= max(||h[n]||, eps);  logit_acc += h[n]·wc  (atomic, /N later)
// ---------------------------------------------------------------------------
__global__ void __launch_bounds__(256)
nshe_normalize_kernel(float* __restrict__ h,
                      const float* __restrict__ wc,
                      float* __restrict__ logit_acc) {
    int row  = blockIdx.x * 8 + (threadIdx.x >> 5);
    int lane = threadIdx.x & 31;
    if (row >= N_NODES) return;

    float* hrow = h + (size_t)row * EDIM;
    float v0 = hrow[lane];
    float v1 = hrow[lane + 32];

    float ss = v0 * v0 + v1 * v1;
#pragma unroll
    for (int m = 16; m >= 1; m >>= 1) ss += __shfl_xor(ss, m, 32);

    float inv = 1.0f / fmaxf(sqrtf(ss), 1e-12f);
    v0 *= inv; v1 *= inv;
    hrow[lane]      = v0;
    hrow[lane + 32] = v1;

    float dotv = v0 * wc[lane] + v1 * wc[lane + 32];
#pragma unroll
    for (int m = 16; m >= 1; m >>= 1) dotv += __shfl_xor(dotv, m, 32);
    if (lane == 0) atomicAdd(logit_acc, dotv);
}

// ---------------------------------------------------------------------------
// 5) Finalize scalar logit
// ---------------------------------------------------------------------------
__global__ void nshe_finalize_kernel(const float* __restrict__ logit_acc,
                                     const float* __restrict__ bc,
                                     float* __restrict__ out) {
    out[0] = logit_acc[0] * (1.0f / (float)N_NODES) + bc[0];
}

extern "C" void kernel_launch(void* const* d_in, const int* in_sizes, int n_in,
                              void* d_out, int out_size, void* d_ws, size_t ws_size,
                              hipStream_t stream) {
    const float* x   = (const float*)d_in[0];
    const float* W   = (const float*)d_in[1];
    const float* b   = (const float*)d_in[2];
    const float* wc  = (const float*)d_in[3];
    const float* bc  = (const float*)d_in[4];
    const int*   src = (const int*)d_in[5];
    const int*   dst = (const int*)d_in[6];

    float* out = (float*)d_out;
    float* h   = out + 1;                            // [N_NODES, 64] (4B-aligned!)
    float* y   = (float*)d_ws;                       // [N_NODES, 64] scratch
    float* logit_acc = y + (size_t)N_NODES * EDIM;   // 1 float scratch

    // 1) agg := b (broadcast), zero logit accumulator
    nshe_init_kernel<<<(N_NODES * EDIM + 255) / 256, 256, 0, stream>>>(h, b, logit_acc);

    // 2) y = x @ W  (WMMA; 100000 rows / 32 rows-per-block = 3125 blocks)
    nshe_gemm_kernel<<<N_NODES / 32, 256, 0, stream>>>(x, W, y);

    // 3) scatter-add over all (edge, dim) pairs
    long long total = (long long)N_EDGES * EDIM;
    nshe_scatter_kernel<<<(unsigned)((total + 255) / 256), 256, 0, stream>>>(y, src, dst, h);

    // 4) normalize rows + accumulate classifier dot (8 rows per block)
    nshe_normalize_kernel<<<(N_NODES + 7) / 8, 256, 0, stream>>>(h, wc, logit_acc);

    // 5) logits = sum/N + bc
    nshe_finalize_kernel<<<1, 1, 0, stream>>>(logit_acc, bc, out);
}